// GenerateActionsBlock_41197326303395
// MI455X (gfx1250) — compile-verified
//
#include <hip/hip_runtime.h>
#include <math.h>
#include <stdint.h>

#define OUT_EMB   128
#define BATCH     32
#define NATOMS    512
#define KV_STRIDE (2 * OUT_EMB + 1)   // 257 floats per (b,n) row
#define LOG_STD_MIN (-20.0f)
#define LOG_STD_MAX (2.0f)
#define HALF_LOG_2PI 0.9189385332046727f
#define INV_SQRT_E  0.08838834764831845f   // 1/sqrt(128)

typedef __attribute__((ext_vector_type(2))) float v2f;
typedef __attribute__((ext_vector_type(8))) float v8f;

#define LDS_STRIDE 132   // floats; %4==0 -> 16B-aligned rows; banks (4r+c)%64 conflict-free
#define NTILES (NATOMS / 16)

// ---- CDNA5 async global->LDS staging (ASYNCcnt-tracked), with safe fallback
#if __has_builtin(__builtin_amdgcn_global_load_async_to_lds_b128)
#define USE_ASYNC_LDS 1
#else
#define USE_ASYNC_LDS 0
#endif

#if USE_ASYNC_LDS
// Parameter types per hipcc diagnostic: pointee is int __attribute__((vector_size(16)))
typedef int v4i_vs __attribute__((vector_size(16)));
typedef __attribute__((address_space(1))) v4i_vs* async_gptr_t;   // global (__device__) src
typedef __attribute__((address_space(3))) v4i_vs* async_lptr_t;   // LDS (__shared__) dst

#if __has_builtin(__builtin_amdgcn_s_wait_asynccnt)
#define WAIT_ASYNC(n) __builtin_amdgcn_s_wait_asynccnt(n)
#else
#define WAIT_ASYNC(n) asm volatile("s_wait_asynccnt %0" ::"i"(n) : "memory")
#endif
#endif

// ---------------------------------------------------------------------------
// Kernel 0: per-batch mask sum into ws[b], and zero the log_prob output slot.
// Runs every call (harness does not re-poison between graph replays).
// ---------------------------------------------------------------------------
__global__ __launch_bounds__(256)
void mask_sum_zero_kernel(const float* __restrict__ mask,
                          float* __restrict__ ws,
                          float* __restrict__ lp_out) {
  __shared__ float red[256];
  const int b = blockIdx.x;
  const int t = threadIdx.x;
  red[t] = mask[b * NATOMS + t] + mask[b * NATOMS + 256 + t];
  __syncthreads();
  for (int off = 128; off > 0; off >>= 1) {
    if (t < off) red[t] += red[t + off];
    __syncthreads();
  }
  if (t == 0) {
    ws[b] = red[0];
    lp_out[b] = 0.0f;
  }
}

// Stage one raw (unmasked) 16x128 V_mu tile: 128 threads, 8/row, 16 floats each.
__device__ __forceinline__
void stage_tile(const float* __restrict__ src_tile,  // = kvb + m_base*KV_STRIDE + OUT_EMB
                float* __restrict__ dst,             // LDS buffer base
                int tid) {
  const int row = tid >> 3;
  const int cb  = (tid & 7) * 16;
  const float* src = src_tile + (size_t)row * KV_STRIDE + cb;
  float* d = dst + row * LDS_STRIDE + cb;
#if USE_ASYNC_LDS
#pragma unroll
  for (int q = 0; q < 4; ++q) {
    __builtin_amdgcn_global_load_async_to_lds_b128(
        (async_gptr_t)(const_cast<float*>(src) + 4 * q),
        (async_lptr_t)(d + 4 * q),
        0, 0);
  }
#else
#pragma unroll
  for (int q = 0; q < 4; ++q)
    *(float4*)(d + 4 * q) = *(const float4*)(src + 4 * q);
#endif
}

// ---------------------------------------------------------------------------
// Main fused kernel. Grid: BATCH*8 blocks of 128 threads (4 waves).
// Block covers 64 n-rows of one batch; wave w owns n-tile [n_base, n_base+16).
// ---------------------------------------------------------------------------
__global__ __launch_bounds__(128)
void gen_actions_kernel(const float* __restrict__ kv,
                        const float* __restrict__ positions,
                        const float* __restrict__ mask,
                        const float* __restrict__ action_scale,
                        const float* __restrict__ eval_actions,
                        const float* __restrict__ mask_sum,
                        float* __restrict__ out) {
#if USE_ASYNC_LDS
  __shared__ float vlds[2][16 * LDS_STRIDE];   // double-buffered V_mu tiles
#else
  __shared__ float vlds[1][16 * LDS_STRIDE];
#endif
  __shared__ float lp_red[4];

  const int tid  = threadIdx.x;
  const int b    = blockIdx.x >> 3;
  const int ngrp = blockIdx.x & 7;
  const int wave = tid >> 5;
  const int lane = tid & 31;
  const int hi   = lane >> 4;        // half-wave select (K split / M split)
  const int lm   = lane & 15;

  const int n_base = ngrp * 64 + wave * 16;

  const float* kvb   = kv        + (size_t)b * NATOMS * KV_STRIDE;
  const float* posb  = positions + (size_t)b * NATOMS * 3;
  const float* maskb = mask      + (size_t)b * NATOMS;
  const float* evalb = eval_actions + (size_t)b * NATOMS * 3;
  const float* vmu0  = kvb + OUT_EMB;            // v_mu = cols [128,256)

  // ---- A fragments: masked K_mu rows of this wave's n-tile, all 32 k-steps.
  // A(16x4) layout: lane holds M=lane%16, VGPR v = K = k0 + 2*hi + v.
  const int   arow  = n_base + lm;
  const float amask = maskb[arow];
  const float* arow_ptr = kvb + (size_t)arow * KV_STRIDE;   // k_mu = cols [0,128)
  v2f afrag[32];
#pragma unroll
  for (int s = 0; s < 32; ++s) {
    const int col = 4 * s + 2 * hi;
    afrag[s].x = arow_ptr[col]     * amask;
    afrag[s].y = arow_ptr[col + 1] * amask;
  }

  // Positions of this lane's 8 C-fragment rows (n = n_base + j + 8*hi).
  float pnx[8], pny[8], pnz[8];
#pragma unroll
  for (int j = 0; j < 8; ++j) {
    const int n = n_base + j + 8 * hi;
    pnx[j] = posb[n * 3 + 0];
    pny[j] = posb[n * 3 + 1];
    pnz[j] = posb[n * 3 + 2];
  }

  float accx[8], accy[8], accz[8];
#pragma unroll
  for (int j = 0; j < 8; ++j) { accx[j] = 0.f; accy[j] = 0.f; accz[j] = 0.f; }

#if USE_ASYNC_LDS
  // Prime the pipeline: async-copy tile 0 into buffer 0.
  stage_tile(vmu0, &vlds[0][0], tid);
#endif

  // ---- m-tile loop -------------------------------------------------------
  for (int mt = 0; mt < NTILES; ++mt) {
    const int m_base = mt * 16;

#if USE_ASYNC_LDS
    // Issue async copy of the NEXT tile, then wait for the CURRENT one
    // (async loads complete in order; keep next tile's 4 ops in flight).
    if (mt + 1 < NTILES) {
      stage_tile(vmu0 + (size_t)(m_base + 16) * KV_STRIDE,
                 &vlds[(mt + 1) & 1][0], tid);
      WAIT_ASYNC(4);
    } else {
      WAIT_ASYNC(0);
    }
    __syncthreads();                       // current tile visible to all waves
    const float* tile = &vlds[mt & 1][0];
#else
    __syncthreads();                       // previous tile fully consumed
    stage_tile(vmu0 + (size_t)m_base * KV_STRIDE, &vlds[0][0], tid);
    __syncthreads();
    const float* tile = &vlds[0][0];
#endif

    // This lane's m column: position + mask (mask folded in post-GEMM).
    const int m = m_base + lm;
    const float pmx = posb[m * 3 + 0];
    const float pmy = posb[m * 3 + 1];
    const float pmz = posb[m * 3 + 2];
    const float mmask = maskb[m];

    // S tile: C = A*B. B(4x16) layout: lane holds N=lane%16, VGPR v = K = k0+2*hi+v.
    v8f cacc = {0.f, 0.f, 0.f, 0.f, 0.f, 0.f, 0.f, 0.f};
    const float* brow = tile + lm * LDS_STRIDE;
#pragma unroll
    for (int s = 0; s < 32; ++s) {
      const int col = 4 * s + 2 * hi;
      v2f bfrag;
      bfrag.x = brow[col];
      bfrag.y = brow[col + 1];
      cacc = __builtin_amdgcn_wmma_f32_16x16x4_f32(
          /*neg_a=*/false, afrag[s],
          /*neg_b=*/false, bfrag,
          /*c_mod=*/(short)0, cacc,
          /*reuse_a=*/false, /*reuse_b=*/false);
    }

    // Fold S * mask_m * unit(pos_n - pos_m) into per-row accumulators.
    const float wm = INV_SQRT_E * mmask;
#pragma unroll
    for (int j = 0; j < 8; ++j) {
      const float s  = cacc[j] * wm;
      const float dx = pnx[j] - pmx;
      const float dy = pny[j] - pmy;
      const float dz = pnz[j] - pmz;
      const float nrm = sqrtf(dx * dx + dy * dy + dz * dz) + 1e-8f;
      const float w = s / nrm;
      accx[j] += w * dx;
      accy[j] += w * dy;
      accz[j] += w * dz;
    }

    __syncthreads();   // all waves done reading before this buffer is re-staged
  }

  // ---- reduce over the 16 lanes of each half-wave (m columns) ------------
#pragma unroll
  for (int off = 1; off < 16; off <<= 1) {
#pragma unroll
    for (int j = 0; j < 8; ++j) {
      accx[j] += __shfl_xor(accx[j], off, 32);
      accy[j] += __shfl_xor(accy[j], off, 32);
      accz[j] += __shfl_xor(accz[j], off, 32);
    }
  }

  // ---- tanh / log-prob / outputs -----------------------------------------
  const float inv_msum = 1.0f / mask_sum[b];
  const float ascale   = action_scale[0];
  float lp = 0.0f;

  if (lm < 3) {
    const int c = lm;   // component handled by this lane
#pragma unroll
    for (int j = 0; j < 8; ++j) {
      const int n = n_base + j + 8 * hi;
      const float mk = maskb[n];
      const float av = (c == 0) ? accx[j] : (c == 1) ? accy[j] : accz[j];
      const float mu = tanhf(av * inv_msum) * ascale;
      const float lsr = kvb[(size_t)n * KV_STRIDE + 2 * OUT_EMB] * mk;
      const float ls  = fminf(fmaxf(lsr, LOG_STD_MIN), LOG_STD_MAX);
      const float sigma = expf(ls) * ascale;
      const float a = evalb[n * 3 + c];
      const float z = (a - mu) / sigma;
      lp += (-0.5f * z * z - logf(sigma) - HALF_LOG_2PI) * mk;
      out[((size_t)b * NATOMS + n) * 3 + c] = a * mk;   // actions output
    }
  }

  // wave-level then block-level log_prob reduction, one atomicAdd per block
#pragma unroll
  for (int off = 1; off < 32; off <<= 1) lp += __shfl_xor(lp, off, 32);
  if (lane == 0) lp_red[wave] = lp;
  __syncthreads();
  if (tid == 0) {
    const float total = lp_red[0] + lp_red[1] + lp_red[2] + lp_red[3];
    atomicAdd(&out[(size_t)BATCH * NATOMS * 3 + b], total);
  }
}

// ---------------------------------------------------------------------------
extern "C" void kernel_launch(void* const* d_in, const int* in_sizes, int n_in,
                              void* d_out, int out_size, void* d_ws, size_t ws_size,
                              hipStream_t stream) {
  const float* kv           = (const float*)d_in[0];
  const float* positions    = (const float*)d_in[1];
  const float* atoms_mask   = (const float*)d_in[2];
  const float* action_scale = (const float*)d_in[3];
  const float* eval_actions = (const float*)d_in[4];
  float* out = (float*)d_out;
  float* ws  = (float*)d_ws;
  float* lp_out = out + (size_t)BATCH * NATOMS * 3;

  mask_sum_zero_kernel<<<BATCH, 256, 0, stream>>>(atoms_mask, ws, lp_out);
  gen_actions_kernel<<<BATCH * 8, 128, 0, stream>>>(
      kv, positions, atoms_mask, action_scale, eval_actions, ws, out);
}